// NMSModule_15006615733118
// MI455X (gfx1250) — compile-verified
//
#include <hip/hip_runtime.h>
#include <hip/hip_bf16.h>
#include <float.h>

// ---- problem constants (match reference) ----
#define B_IMG     32
#define N_BOX     262144
#define MAX_DET   100
#define IOU_THR   0.5f
#define SCORE_THR 0.05f

// ---- tiling config ----
#define NTHREADS        1024                  // 32 wave32s per block
#define TILE            4096                  // boxes per LDS tile
#define NTILES          (N_BOX / TILE)        // 64
#define BOX_PER_THREAD  (TILE / NTHREADS)     // 4
#define ALIVE_BITS      (N_BOX / NTHREADS)    // 256 bits per thread
#define ALIVE_WORDS     (ALIVE_BITS / 32)     // 8 u32 per thread

typedef unsigned int       u32;
typedef unsigned long long u64;
typedef u32 v4u __attribute__((ext_vector_type(4)));
typedef int v4i __attribute__((ext_vector_type(4)));
typedef int v8i __attribute__((ext_vector_type(8)));

// ---- Tensor Data Mover: 1-D contiguous copy of `nelem` f32 from global to LDS ----
// D# group0: [1:0]=count=1, [63:32]=lds_addr, [120:64]=global_addr, [127:126]=type=2
// D# group1: data_size=4B (bits 17:16 = 2), tensor_dim0 = tile_dim0 = nelem (tile fully
//            in-bounds by construction, so no OOB handling is exercised).
__device__ __forceinline__ void tdm_load_1d(u32 lds_byte_off, const void* gptr, u32 nelem) {
  u64 ga = (u64)(uintptr_t)gptr;
  v4u g0;
  g0[0] = 1u;                                          // count=1, user descriptor
  g0[1] = lds_byte_off;                                // lds_addr
  g0[2] = (u32)ga;                                     // global_addr[31:0]
  g0[3] = (u32)((ga >> 32) & 0x1FFFFFFu) | (2u << 30); // global_addr[56:32] | type=2
  v8i g1;
  g1[0] = 0x00020000;                                  // data_size = 4 bytes
  g1[1] = (int)((nelem & 0xFFFFu) << 16);              // tensor_dim0[15:0]
  g1[2] = (int)((nelem >> 16) & 0xFFFFu);              // tensor_dim0[31:16]
  g1[3] = (int)((nelem & 0xFFFFu) << 16);              // tile_dim0 (<= 65535)
  g1[4] = 0; g1[5] = 0; g1[6] = 0; g1[7] = 0;          // tile_dim1/2 unused, strides 0
  v4i z4 = {0, 0, 0, 0};
#if __clang_major__ >= 23
  v8i z8 = {0, 0, 0, 0, 0, 0, 0, 0};
  __builtin_amdgcn_tensor_load_to_lds(g0, g1, z4, z4, z8, 0);
#else
  __builtin_amdgcn_tensor_load_to_lds(g0, g1, z4, z4, 0);
#endif
}

__global__ __launch_bounds__(NTHREADS)
void nms_kernel(const float* __restrict__ boxes,
                const float* __restrict__ scores,
                const int*   __restrict__ labels,
                float* __restrict__ outBoxes,   // [B, MAX_DET, 4]
                float* __restrict__ outScores,  // [B, MAX_DET]
                float* __restrict__ outLabels)  // [B, MAX_DET]
{
  // 2 x 64 KB box tiles + 2 x 16 KB score tiles = 160 KB LDS total.
  __shared__ float sBoxes[2][TILE * 4];
  __shared__ float sScores[2][TILE];

  // Reduction scratch + pick broadcast alias into sScores[1]; they are only
  // live between the end of the tile loop and the top-of-sweep barrier, a
  // window in which no DMA targets that buffer.
  float* const sRedS = &sScores[1][0];          // 32 floats
  int*   const sRedI = (int*)&sScores[1][64];   // 32 ints
  float* const sPick = &sScores[1][128];        // x1,y1,x2,y2,area,valid

  const int img  = blockIdx.x;
  const int t    = threadIdx.x;
  const int wave = t >> 5;
  const int lane = t & 31;

  const float* gBoxes  = boxes  + (size_t)img * N_BOX * 4;
  const float* gScores = scores + (size_t)img * N_BOX;
  const int*   gLabels = labels + (size_t)img * N_BOX;

  // LDS byte offsets for the TDM descriptors (low 32 bits of the flat address
  // are the LDS offset per the aperture scheme).
  const u32 boxOff[2] = { (u32)(uintptr_t)(void*)&sBoxes[0][0],
                          (u32)(uintptr_t)(void*)&sBoxes[1][0] };
  const u32 scoOff[2] = { (u32)(uintptr_t)(void*)&sScores[0][0],
                          (u32)(uintptr_t)(void*)&sScores[1][0] };

  u32 alive[ALIVE_WORDS];   // fully initialized during sweep 0

  if (t == 0) sPick[5] = 0.0f;   // no pick yet -> sweep 0 does no suppression
  __syncthreads();

  for (int k = 0; k < MAX_DET; ++k) {
    // Broadcast previous pick into registers.
    const float px1 = sPick[0], py1 = sPick[1], px2 = sPick[2], py2 = sPick[3];
    const float parea  = sPick[4];
    const bool  pvalid = (sPick[5] != 0.0f);
    __syncthreads();   // everyone has the pick before DMA may overwrite the area

    // Prime the double buffer: tiles 0 and 1 (2 TDM ops per tile).
    if (wave == 0) {
      tdm_load_1d(boxOff[0], gBoxes,                    TILE * 4);
      tdm_load_1d(scoOff[0], gScores,                   TILE);
      tdm_load_1d(boxOff[1], gBoxes + (size_t)TILE * 4, TILE * 4);
      tdm_load_1d(scoOff[1], gScores + TILE,            TILE);
    }

    float bestS = -FLT_MAX;
    int   bestI = 0;

    for (int tile = 0; tile < NTILES; ++tile) {
      const int cur = tile & 1;
      if (wave == 0) {
        // tile's 2 ops done; (tile+1)'s 2 ops may remain in flight
        if (tile + 1 < NTILES) __builtin_amdgcn_s_wait_tensorcnt((short)2);
        else                   __builtin_amdgcn_s_wait_tensorcnt((short)0);
      }
      __syncthreads();   // buffer `cur` is ready for everyone

#pragma unroll
      for (int s = 0; s < BOX_PER_THREAD; ++s) {
        const int local  = s * NTHREADS + t;
        const int gidx   = tile * TILE + local;
        const int bitIdx = tile * BOX_PER_THREAD + s;
        const u32 w = (u32)bitIdx >> 5;
        const u32 m = 1u << (bitIdx & 31);

        const float sc = sScores[cur][local];
        bool al = (k == 0) ? (sc > SCORE_THR) : ((alive[w] & m) != 0u);
        if (al) {
          const float4 bx = *(const float4*)&sBoxes[cur][local * 4];
          if (pvalid) {
            const float xx1 = fmaxf(px1, bx.x), yy1 = fmaxf(py1, bx.y);
            const float xx2 = fminf(px2, bx.z), yy2 = fminf(py2, bx.w);
            const float inter = fmaxf(xx2 - xx1, 0.0f) * fmaxf(yy2 - yy1, 0.0f);
            const float area  = (bx.z - bx.x) * (bx.w - bx.y);
            const float uni   = parea + area - inter;
            if (inter > IOU_THR * uni) al = false;   // iou > thr  (uni > 0 always)
          }
          if (al && sc > bestS) { bestS = sc; bestI = gidx; }
        }
        if (al) alive[w] |= m; else alive[w] &= ~m;
      }

      __syncthreads();   // everyone done reading buffer `cur`
      if (wave == 0 && tile + 2 < NTILES) {
        tdm_load_1d(boxOff[cur], gBoxes + (size_t)(tile + 2) * TILE * 4, TILE * 4);
        tdm_load_1d(scoOff[cur], gScores + (size_t)(tile + 2) * TILE,    TILE);
      }
    }

    // ---- block-wide argmax (tie-break toward lower index, like jnp.argmax) ----
    for (int off = 16; off > 0; off >>= 1) {
      const float os = __shfl_down(bestS, off, 32);
      const int   oi = __shfl_down(bestI, off, 32);
      if (os > bestS || (os == bestS && oi < bestI)) { bestS = os; bestI = oi; }
    }
    if (lane == 0) { sRedS[wave] = bestS; sRedI[wave] = bestI; }
    __syncthreads();

    if (wave == 0) {
      float rs = sRedS[lane];
      int   ri = sRedI[lane];
      for (int off = 16; off > 0; off >>= 1) {
        const float os = __shfl_down(rs, off, 32);
        const int   oi = __shfl_down(ri, off, 32);
        if (os > rs || (os == rs && oi < ri)) { rs = os; ri = oi; }
      }
      if (lane == 0) {
        const int o = img * MAX_DET + k;
        if (rs > -FLT_MAX) {   // a live box existed
          const float bx1 = gBoxes[(size_t)ri * 4 + 0];
          const float by1 = gBoxes[(size_t)ri * 4 + 1];
          const float bx2 = gBoxes[(size_t)ri * 4 + 2];
          const float by2 = gBoxes[(size_t)ri * 4 + 3];
          outBoxes[(size_t)o * 4 + 0] = bx1;
          outBoxes[(size_t)o * 4 + 1] = by1;
          outBoxes[(size_t)o * 4 + 2] = bx2;
          outBoxes[(size_t)o * 4 + 3] = by2;
          outScores[o] = rs;
          outLabels[o] = (float)gLabels[ri];
          sPick[0] = bx1; sPick[1] = by1; sPick[2] = bx2; sPick[3] = by2;
          sPick[4] = (bx2 - bx1) * (by2 - by1);
          sPick[5] = 1.0f;
        } else {
          outBoxes[(size_t)o * 4 + 0] = 0.0f;
          outBoxes[(size_t)o * 4 + 1] = 0.0f;
          outBoxes[(size_t)o * 4 + 2] = 0.0f;
          outBoxes[(size_t)o * 4 + 3] = 0.0f;
          outScores[o] = 0.0f;
          outLabels[o] = 0.0f;
          sPick[5] = 0.0f;
        }
      }
    }
    __syncthreads();   // pick published before next sweep reads it
  }
}

extern "C" void kernel_launch(void* const* d_in, const int* in_sizes, int n_in,
                              void* d_out, int out_size, void* d_ws, size_t ws_size,
                              hipStream_t stream) {
  (void)in_sizes; (void)n_in; (void)out_size; (void)d_ws; (void)ws_size;
  const float* boxes  = (const float*)d_in[0];
  const float* scores = (const float*)d_in[1];
  const int*   labels = (const int*)d_in[2];

  float* out       = (float*)d_out;
  float* outBoxes  = out;                               // 32*100*4
  float* outScores = out + (size_t)B_IMG * MAX_DET * 4; // 32*100
  float* outLabels = out + (size_t)B_IMG * MAX_DET * 5; // 32*100

  nms_kernel<<<B_IMG, NTHREADS, 0, stream>>>(boxes, scores, labels,
                                             outBoxes, outScores, outLabels);
}